// DynamicAdjacency_79697413144624
// MI455X (gfx1250) — compile-verified
//
#include <hip/hip_runtime.h>
#include <hip/hip_bf16.h>
#include <math.h>

typedef __attribute__((ext_vector_type(2))) float v2f;
typedef __attribute__((ext_vector_type(8))) float v8f;
typedef __attribute__((ext_vector_type(4))) int   v4i;

#define N 1024
#define H 128
#define PAD 132   // floats; 528B rows: 16B-aligned for B128, 4-bank skew per row

typedef __attribute__((address_space(1))) v4i g_v4i;   // global int4
typedef __attribute__((address_space(3))) v4i l_v4i;   // LDS int4

// One 16-byte async copy global -> LDS (ASYNCcnt-tracked, bypasses VGPRs).
__device__ __forceinline__ void async_copy16(const float* g, float* l) {
    __builtin_amdgcn_global_load_async_to_lds_b128(
        (g_v4i*)g, (l_v4i*)l, /*offset=*/0, /*cpol=*/0);
}

// ---------------------------------------------------------------------------
// Kernel 1: row-wise L2 normalize.  One 128-thread block per row.
// ---------------------------------------------------------------------------
__global__ void k_normalize(const float* __restrict__ x, float* __restrict__ e) {
    __shared__ float red[H];
    const int row = blockIdx.x;
    const int t   = threadIdx.x;           // 0..127
    const float v = x[row * H + t];
    red[t] = v * v;
    __syncthreads();
    for (int s = H / 2; s > 0; s >>= 1) {
        if (t < s) red[t] += red[t + s];
        __syncthreads();
    }
    const float nrm = sqrtf(red[0]);
    e[row * H + t] = v / fmaxf(nrm, 1e-12f);
}

// ---------------------------------------------------------------------------
// Kernel 2: fused GEMM  LR = e @ [W1[:H] | W1[H:]]   (1024x128)x(128x256),
// fp32 WMMA (V_WMMA_F32_16X16X4_F32), one wave per 16x16 tile, K stepped by 4.
// b1 is folded into the left half here so kernel 3 staging is a pure copy.
// ---------------------------------------------------------------------------
__global__ void k_gemm_wmma(const float* __restrict__ e,
                            const float* __restrict__ W1,
                            const float* __restrict__ b1,
                            float* __restrict__ LR) {
    const int lane = threadIdx.x;          // 0..31
    const int half = lane >> 4;            // 0 or 1
    const int l15  = lane & 15;
    const int row  = blockIdx.x * 16 + l15;       // A row this lane feeds
    const int n    = blockIdx.y * 16 + l15;       // virtual B column (0..255)
    const int koff = half * 2;

    const float* bcol = (n < H) ? (W1 + n) : (W1 + H * H + (n - H));
    const float* arow = e + row * H;

    v8f acc = {};
    #pragma unroll 4
    for (int k0 = 0; k0 < H; k0 += 4) {
        v2f a, b;
        a.x = arow[k0 + koff + 0];
        a.y = arow[k0 + koff + 1];
        b.x = bcol[(k0 + koff + 0) * H];
        b.y = bcol[(k0 + koff + 1) * H];
        acc = __builtin_amdgcn_wmma_f32_16x16x4_f32(
            false, a, false, b, (short)0, acc, false, false);
    }

    const float bias   = (n < H) ? b1[n] : 0.0f;  // fold b1 into left half
    const int   outCol = blockIdx.y * 16 + l15;
    #pragma unroll
    for (int r = 0; r < 8; ++r) {
        const int outRow = blockIdx.x * 16 + half * 8 + r;
        LR[outRow * 256 + outCol] = acc[r] + bias;
    }
}

// ---------------------------------------------------------------------------
// Kernel 3: pairwise relu-dot tile kernel.  32x32 output tile per 256-thread
// block.  L/R tiles staged via GLOBAL_LOAD_ASYNC_TO_LDS_B128 (ASYNCcnt),
// rows padded to 132 floats (16B-aligned, conflict-free reads).
// Micro-tile per thread: rows {ti, ti+16} x cols {tj, tj+16}.
// ---------------------------------------------------------------------------
__global__ void k_pair_logits(const float* __restrict__ LR,
                              const float* __restrict__ W2,
                              const float* __restrict__ b2,
                              float* __restrict__ out) {
    __shared__ __align__(16) float Lt[32][PAD];
    __shared__ __align__(16) float Rt[32][PAD];
    __shared__ float w2s[H];

    const int i0 = blockIdx.y * 32;
    const int j0 = blockIdx.x * 32;
    const int t  = threadIdx.x;            // 0..255

    // 32 rows x 32 float4-chunks per tile = 1024 chunks; 4 per thread per tile.
    for (int idx = t; idx < 32 * 32; idx += 256) {
        const int r  = idx >> 5;
        const int c4 = (idx & 31) * 4;
        async_copy16((float*)&LR[(i0 + r) * 256 + c4],     &Lt[r][c4]);  // left+b1
        async_copy16((float*)&LR[(j0 + r) * 256 + H + c4], &Rt[r][c4]);  // right
    }
    if (t < H) w2s[t] = W2[t];
    asm volatile("s_wait_asynccnt 0x0" ::: "memory");
    __syncthreads();

    const int ti = t >> 4;                 // 0..15
    const int tj = t & 15;                 // 0..15
    float a00 = 0.f, a01 = 0.f, a10 = 0.f, a11 = 0.f;

    #pragma unroll 4
    for (int h = 0; h < H; ++h) {
        const float w  = w2s[h];
        const float l0 = Lt[ti][h];
        const float l1 = Lt[ti + 16][h];
        const float r0 = Rt[tj][h];
        const float r1 = Rt[tj + 16][h];
        a00 += fmaxf(l0 + r0, 0.f) * w;
        a01 += fmaxf(l0 + r1, 0.f) * w;
        a10 += fmaxf(l1 + r0, 0.f) * w;
        a11 += fmaxf(l1 + r1, 0.f) * w;
    }

    const float bb = b2[0];
    out[(i0 + ti +  0) * N + (j0 + tj +  0)] = a00 + bb;
    out[(i0 + ti +  0) * N + (j0 + tj + 16)] = a01 + bb;
    out[(i0 + ti + 16) * N + (j0 + tj +  0)] = a10 + bb;
    out[(i0 + ti + 16) * N + (j0 + tj + 16)] = a11 + bb;
}

// ---------------------------------------------------------------------------
// Kernel 4: in-place row softmax, one 256-thread block per row.
// ---------------------------------------------------------------------------
__global__ void k_softmax_rows(float* __restrict__ out) {
    __shared__ float red[256];
    const int row = blockIdx.x;
    const int t   = threadIdx.x;
    float4* rowp  = (float4*)(out + row * N);
    float4 v = rowp[t];

    float m = fmaxf(fmaxf(v.x, v.y), fmaxf(v.z, v.w));
    red[t] = m;
    __syncthreads();
    for (int s = 128; s > 0; s >>= 1) {
        if (t < s) red[t] = fmaxf(red[t], red[t + s]);
        __syncthreads();
    }
    const float M = red[0];
    __syncthreads();

    v.x = __expf(v.x - M);
    v.y = __expf(v.y - M);
    v.z = __expf(v.z - M);
    v.w = __expf(v.w - M);
    red[t] = v.x + v.y + v.z + v.w;
    __syncthreads();
    for (int s = 128; s > 0; s >>= 1) {
        if (t < s) red[t] += red[t + s];
        __syncthreads();
    }
    const float inv = 1.0f / red[0];

    v.x *= inv; v.y *= inv; v.z *= inv; v.w *= inv;
    rowp[t] = v;
}

// ---------------------------------------------------------------------------
extern "C" void kernel_launch(void* const* d_in, const int* in_sizes, int n_in,
                              void* d_out, int out_size, void* d_ws, size_t ws_size,
                              hipStream_t stream) {
    const float* node_emb = (const float*)d_in[0];   // (1024,128)
    const float* W1       = (const float*)d_in[1];   // (256,128)
    const float* b1       = (const float*)d_in[2];   // (128,)
    const float* W2       = (const float*)d_in[3];   // (128,1)
    const float* b2       = (const float*)d_in[4];   // (1,)
    float*       out      = (float*)d_out;           // (1024,1024)

    float* e  = (float*)d_ws;        // 1024*128 floats (512 KB)
    float* LR = e + N * H;           // 1024*256 floats (1 MB): [left+b1 | right]

    k_normalize<<<N, H, 0, stream>>>(node_emb, e);
    k_gemm_wmma<<<dim3(N / 16, 256 / 16), 32, 0, stream>>>(e, W1, b1, LR);
    k_pair_logits<<<dim3(N / 32, N / 32), 256, 0, stream>>>(LR, W2, b2, out);
    k_softmax_rows<<<N, 256, 0, stream>>>(out);
}